// SKA_75453985457140
// MI455X (gfx1250) — compile-verified
//
#include <hip/hip_runtime.h>

// Problem constants (from the reference):
//   x  : (8, 256, 64, 64)  f32
//   w  : (8,  32,  9, 64, 64) f32, channel c uses weight plane (c % 32)
//   out: (8, 256, 64, 64)  f32
#define B_   8
#define C_   256
#define H_   64
#define W_   64
#define HW_  4096           // 64*64
#define WC_  32
#define K2_  9

typedef __attribute__((ext_vector_type(4))) float v4f;
typedef __attribute__((ext_vector_type(4))) int   v4i;

#if defined(__AMDGCN__) && __has_builtin(__builtin_amdgcn_global_load_async_to_lds_b128)
#define HAVE_ASYNC_LDS 1
typedef __attribute__((address_space(1))) v4i* as1_v4i_ptr;
typedef __attribute__((address_space(3))) v4i* as3_v4i_ptr;
#else
#define HAVE_ASYNC_LDS 0
#endif

__global__ __launch_bounds__(256) void ska_involution3x3(
    const float* __restrict__ x,
    const float* __restrict__ w,
    float* __restrict__ out)
{
  // Tile layout: tile row (r+1) holds x row r; rows 0 and 65 are the zero halo.
  // Stride 64 floats => the interior copy is contiguous with a +64-float offset.
  __shared__ alignas(16) float tile[66 * W_];

  const int bc = blockIdx.x;          // b * 256 + c
  const int b  = bc >> 8;
  const int c  = bc & (C_ - 1);
  const int t  = threadIdx.x;

  const float* xp = x + (size_t)bc * HW_;
  const float* wp = w + (size_t)((b * WC_ + (c & (WC_ - 1))) * K2_) * HW_;
  float*       op = out + (size_t)bc * HW_;

  // Zero the top/bottom halo rows (128 floats).
  if (t < 2 * W_) {
    const int row = (t < W_) ? 0 : 65;
    tile[row * W_ + (t & (W_ - 1))] = 0.0f;
  }

  // Stage the 64x64 plane into LDS. 4096 dwords = 1024 x b128 transfers,
  // 4 per thread. Both global and LDS addresses are 16B aligned.
#if HAVE_ASYNC_LDS
  {
    const float* gsrc = xp + t * 4;
    float*       ldst = &tile[W_ + t * 4];
#pragma unroll
    for (int i = 0; i < 4; ++i) {
      __builtin_amdgcn_global_load_async_to_lds_b128(
          (as1_v4i_ptr)(gsrc + i * 1024),
          (as3_v4i_ptr)(ldst + i * 1024),
          /*offset=*/0, /*cpol=*/0);
    }
#if __has_builtin(__builtin_amdgcn_s_wait_asynccnt)
    __builtin_amdgcn_s_wait_asynccnt(0);
#else
    asm volatile("s_wait_asynccnt 0" ::: "memory");
#endif
  }
#else
#pragma unroll
  for (int i = 0; i < 4; ++i) {
    *(v4f*)&tile[W_ + t * 4 + i * 1024] = *(const v4f*)(xp + t * 4 + i * 1024);
  }
#endif
  __syncthreads();

  // Each thread computes 4 consecutive pixels per iteration, 4 iterations.
#pragma unroll
  for (int it = 0; it < 4; ++it) {
    const int p0  = t * 4 + it * 1024;   // flat pixel index, multiple of 4
    const int r   = p0 >> 6;             // output row 0..63
    const int col = p0 & (W_ - 1);       // output col, multiple of 4

    // a[dr][idx] = x[r + dr - 1, col - 1 + idx]  (zeros at the borders)
    float a[3][6];
#pragma unroll
    for (int dr = 0; dr < 3; ++dr) {
      const float* rowp = &tile[(r + dr) * W_];   // tile row = x row (r+dr-1) + 1
      a[dr][0] = (col == 0) ? 0.0f : rowp[col - 1];
      const v4f m = *(const v4f*)&rowp[col];
      a[dr][1] = m.x; a[dr][2] = m.y; a[dr][3] = m.z; a[dr][4] = m.w;
      a[dr][5] = (col == W_ - 4) ? 0.0f : rowp[col + 4];
    }

    v4f acc = {0.0f, 0.0f, 0.0f, 0.0f};
#pragma unroll
    for (int k = 0; k < K2_; ++k) {
      const int dr = k / 3;
      const int dj = k % 3;
      const v4f wk = *(const v4f*)(wp + k * HW_ + p0);   // coalesced b128
      acc.x = __builtin_fmaf(a[dr][dj + 0], wk.x, acc.x);
      acc.y = __builtin_fmaf(a[dr][dj + 1], wk.y, acc.y);
      acc.z = __builtin_fmaf(a[dr][dj + 2], wk.z, acc.z);
      acc.w = __builtin_fmaf(a[dr][dj + 3], wk.w, acc.w);
    }
    *(v4f*)(op + p0) = acc;
  }
}

extern "C" void kernel_launch(void* const* d_in, const int* in_sizes, int n_in,
                              void* d_out, int out_size, void* d_ws, size_t ws_size,
                              hipStream_t stream) {
  (void)in_sizes; (void)n_in; (void)d_ws; (void)ws_size; (void)out_size;
  const float* x = (const float*)d_in[0];
  const float* w = (const float*)d_in[1];
  float*     out = (float*)d_out;
  dim3 grid(B_ * C_);
  dim3 block(256);
  hipLaunchKernelGGL(ska_involution3x3, grid, block, 0, stream, x, w, out);
}